// DecisionGate_74062416052252
// MI455X (gfx1250) — compile-verified
//
#include <hip/hip_runtime.h>
#include <stdint.h>

// DecisionGate on MI455X (gfx1250, wave32).
// Store-bandwidth-bound: ~537 MB of streaming f32 output vs ~9 MB of input
// (floor ~23 us @ 23.3 TB/s HBM). CDNA5-specific paths used:
//   * TDM tensor_load_to_lds DMA of the act tile (TENSORcnt, overlapped with
//     gate math) -- confirmed lowering to `tensor_load_to_lds s[..], s[..]`.
//   * s_wait_tensorcnt 0 before the workgroup barrier.
//   * non-temporal global_store_b128 for the never-re-read dispatch tensor,
//     keeping x/act resident in the 192 MB L2.
// WMMA is deliberately NOT used: the dispatch is a K=1 rank-1 outer product;
// 16x16x4 f32 WMMA would run at 12.5% MAC utilization and its C/D lane layout
// would break the 512 B/instruction contiguous wave32 store stream that is
// the real bottleneck.

typedef float    f32x4 __attribute__((ext_vector_type(4)));
typedef unsigned u32x4 __attribute__((ext_vector_type(4)));
typedef int      i32x4 __attribute__((ext_vector_type(4)));
typedef int      i32x8 __attribute__((ext_vector_type(8)));

#define P_NODES   64
#define D_DIM     512
#define ROWS      4          // batch rows per workgroup
#define NTHREADS  256        // 8 wave32 waves

__global__ __launch_bounds__(NTHREADS)
void decision_gate_kernel(const float* __restrict__ x,
                          const float* __restrict__ act,
                          float* __restrict__ out_g,
                          float* __restrict__ out_mask,
                          float* __restrict__ out_disp,
                          int nrows) {
    __shared__ float act_s[ROWS * D_DIM];   // 8 KB staged act tile
    __shared__ float w_s[ROWS * P_NODES];   // 1 KB gate weights

    const int tid = threadIdx.x;
    const int b0  = blockIdx.x * ROWS;
    const bool full_tile = (b0 + ROWS <= nrows);

    // ---------- Phase A: DMA act[b0 : b0+ROWS, :] into LDS via the TDM ----------
#if __has_builtin(__builtin_amdgcn_tensor_load_to_lds)
    if (tid < 32) {  // one wave issues the descriptor (TDM ignores EXEC anyway)
        const uint32_t lds_addr = (uint32_t)(uintptr_t)(void*)act_s;
        const uint64_t gaddr    = (uint64_t)(uintptr_t)(act + (size_t)b0 * D_DIM);
        const uint32_t rem_rows = (uint32_t)(nrows - b0);   // tensor_dim1 for OOB

        // D# group 0 (ISA 08 §8.3): count=1 | lds_addr | global_addr | type=2
        u32x4 g0;
        g0.x = 1u;
        g0.y = lds_addr;
        g0.z = (uint32_t)(gaddr & 0xFFFFFFFFu);
        g0.w = (uint32_t)((gaddr >> 32) & 0x1FFFFFFu) | (2u << 30);

        // D# group 1 (§8.4): data_size=2 (4B), tensor_dim0=512, tensor_dim1=rem,
        // tile_dim0=512, tile_dim1=ROWS, tensor_dim0_stride=512
        i32x8 g1;
        g1[0] = (int)(2u << 16);                                   // data_size
        g1[1] = (int)((uint32_t)D_DIM << 16);                      // tensor_dim0 lo16 @bit48
        g1[2] = (int)((rem_rows & 0xFFFFu) << 16);                 // tensor_dim1 lo16 @bit80
        g1[3] = (int)((rem_rows >> 16) | ((uint32_t)D_DIM << 16)); // dim1 hi | tile_dim0
        g1[4] = (int)ROWS;                                         // tile_dim1 (tile_dim2=0)
        g1[5] = (int)D_DIM;                                        // tensor_dim0_stride lo32
        g1[6] = 0;
        g1[7] = 0;

        i32x4 gz = {0, 0, 0, 0};                                   // 2D: groups 2/3 -> NULL
#if defined(__clang_major__) && (__clang_major__ >= 23)
        i32x8 gz8 = {0, 0, 0, 0, 0, 0, 0, 0};
        __builtin_amdgcn_tensor_load_to_lds(g0, g1, gz, gz, gz8, 0);
#else
        __builtin_amdgcn_tensor_load_to_lds(g0, g1, gz, gz, 0);
#endif
    }
#else
    // Fallback: cooperative VALU staging (b128 loads, b128 LDS stores)
    for (int i = tid; i < ROWS * D_DIM / 4; i += NTHREADS) {
        ((f32x4*)act_s)[i] = ((const f32x4*)(act + (size_t)b0 * D_DIM))[i];
    }
#endif

    // ---------- Phase B (overlaps the DMA): gate math, g/mask output ----------
    {
        const int r = tid >> 6;            // row within tile
        const int p = tid & (P_NODES - 1); // node
        const int b = b0 + r;
        float w = 0.0f;
        if (b < nrows) {
            const float xv = x[(size_t)b * P_NODES + p];
            const float x2 = xv * xv;                 // |x|^4 == x^4
            const float g  = 1.0f / (1.0f + x2 * x2); // soft gate, A=1, 2B=4
            const bool  m  = (g >= 0.5f);
            out_g   [(size_t)b * P_NODES + p] = g;
            out_mask[(size_t)b * P_NODES + p] = m ? 1.0f : 0.0f;
            w = m ? g : 0.0f;
        }
        w_s[tid] = w;
    }

#if __has_builtin(__builtin_amdgcn_tensor_load_to_lds)
#if __has_builtin(__builtin_amdgcn_s_wait_tensorcnt)
    __builtin_amdgcn_s_wait_tensorcnt(0);
#else
    asm volatile("s_wait_tensorcnt 0" ::: "memory");
#endif
#endif
    __syncthreads();

    // ---------- Phase C: stream outer products with NT b128 stores ----------
    // 4 rows * 64 nodes * 512 f32 = 512 KB per WG; 128 consecutive lanes write
    // one contiguous 2 KB (r,p)-row segment -> maximally coalesced wave32 stores.
    const f32x4* __restrict__ act4 = (const f32x4*)act_s;
    f32x4* __restrict__ disp4 = (f32x4*)(out_disp + (size_t)b0 * P_NODES * D_DIM);
    constexpr int TOT4 = ROWS * P_NODES * (D_DIM / 4);  // 32768 float4s

    if (full_tile) {
        // Hot path: pure store stream, no per-iteration bounds logic.
#pragma unroll 4
        for (int i = tid; i < TOT4; i += NTHREADS) {
            const int d4 = i & (D_DIM / 4 - 1);
            const int rp = i >> 7;                              // r*64 + p
            const float w = w_s[rp];                            // LDS broadcast
            const f32x4 a = act4[(rp >> 6) * (D_DIM / 4) + d4]; // conflict-free b128
            const f32x4 o = a * w;
            __builtin_nontemporal_store(o, &disp4[(size_t)rp * (D_DIM / 4) + d4]);
        }
    } else {
        // Ragged tail tile (not hit for B % ROWS == 0): guarded variant.
        for (int i = tid; i < TOT4; i += NTHREADS) {
            const int d4 = i & (D_DIM / 4 - 1);
            const int rp = i >> 7;
            if (b0 + (rp >> 6) < nrows) {
                const float w = w_s[rp];
                const f32x4 a = act4[(rp >> 6) * (D_DIM / 4) + d4];
                const f32x4 o = a * w;
                __builtin_nontemporal_store(o, &disp4[(size_t)rp * (D_DIM / 4) + d4]);
            }
        }
    }
}

extern "C" void kernel_launch(void* const* d_in, const int* in_sizes, int n_in,
                              void* d_out, int out_size, void* d_ws, size_t ws_size,
                              hipStream_t stream) {
    const float* x   = (const float*)d_in[0];   // [B, 64] f32
    const float* act = (const float*)d_in[1];   // [B, 512] f32
    // d_in[2] = batch_inds: identity arange, unused by the dense formulation.

    const int B = in_sizes[1] / D_DIM;

    float* out_g    = (float*)d_out;                        // [B, 64]
    float* out_mask = out_g    + (size_t)B * P_NODES;       // [B, 64] (bool as 0/1)
    float* out_disp = out_mask + (size_t)B * P_NODES;       // [B, 64, 512]

    const int grid = (B + ROWS - 1) / ROWS;
    decision_gate_kernel<<<grid, NTHREADS, 0, stream>>>(x, act, out_g, out_mask,
                                                        out_disp, B);
}